// EpisodicMemoryBank_56745107914974
// MI455X (gfx1250) — compile-verified
//
#include <hip/hip_runtime.h>
#include <cstdint>
#include <cstddef>

// EpisodicMemoryBank store(): masked FIFO circular-buffer scatter.
// Inputs (setup_inputs order):
//  0 tokens            (N*D)   f32
//  1 memory_tokens     (C*D)   f32
//  2 object_slot_ids   (N)     i32
//  3 visibility_mask   (N)     bool -> int
//  4 valid_mask        (N)     bool -> int
//  5 object_ids        (C)     i32
//  6 frame_ids_buf     (C)     i32
//  7 valid_buf         (C)     bool -> int
//  8 frame_id          (1)     i32
//  9 write_ptr         (1)     i32
// Outputs (concatenated, float32): mem (C*D), oid (C), fid (C), val (C), new_ptr (1)

#define CAP   262144
#define DM    1024
#define NTOK  131072
#define RPB   4          // rows per block in the big copy

// ---------------------------------------------------------------- init inv map
__global__ void init_inv(int* __restrict__ inv) {
    int i = blockIdx.x * blockDim.x + threadIdx.x;
    if (i < CAP) inv[i] = -1;
}

// ------------------------------------------------------- per-block mask counts
__global__ void count_mask(const int* __restrict__ vis, const int* __restrict__ val,
                           int* __restrict__ blockCounts) {
    __shared__ int red[256];
    const int t = threadIdx.x;
    const int i = blockIdx.x * 256 + t;
    int m = (vis[i] != 0) && (val[i] != 0);
    red[t] = m;
    __syncthreads();
#pragma unroll
    for (int s = 128; s > 0; s >>= 1) {
        if (t < s) red[t] += red[t + s];
        __syncthreads();
    }
    if (t == 0) blockCounts[blockIdx.x] = red[0];
}

// ---------------------------------------- scan 512 block counts (single block)
__global__ void scan_block_counts(const int* __restrict__ counts, int* __restrict__ offsets,
                                  const int* __restrict__ wp, float* __restrict__ out_newptr) {
    __shared__ int tmp[512];
    const int t = threadIdx.x;
    const int own = counts[t];
    tmp[t] = own;
    __syncthreads();
    for (int off = 1; off < 512; off <<= 1) {
        int v = (t >= off) ? tmp[t - off] : 0;
        __syncthreads();
        tmp[t] += v;
        __syncthreads();
    }
    offsets[t] = tmp[t] - own;                 // exclusive offset for block t
    if (t == 511) {
        int total = tmp[511];
        out_newptr[0] = (float)((wp[0] + total) % CAP);
    }
}

// ---------------------------- per-token rank -> unique slot -> inverse mapping
__global__ void scatter_inv(const int* __restrict__ vis, const int* __restrict__ val,
                            const int* __restrict__ blockOffsets, const int* __restrict__ wp,
                            int* __restrict__ inv) {
    __shared__ int tmp[256];
    const int t = threadIdx.x;
    const int i = blockIdx.x * 256 + t;
    const int m = (vis[i] != 0) && (val[i] != 0);
    tmp[t] = m;
    __syncthreads();
    for (int off = 1; off < 256; off <<= 1) {
        int v = (t >= off) ? tmp[t - off] : 0;
        __syncthreads();
        tmp[t] += v;
        __syncthreads();
    }
    if (m) {
        int rank = blockOffsets[blockIdx.x] + (tmp[t] - 1);  // global rank among stored
        int pos = (wp[0] + rank) % CAP;                      // unique slot
        inv[pos] = i;
    }
}

// -------------------------------------------------- metadata outputs (gather)
__global__ void small_outputs(const int* __restrict__ inv, const int* __restrict__ slot_ids,
                              const int* __restrict__ obj_ids, const int* __restrict__ fid_buf,
                              const int* __restrict__ val_buf, const int* __restrict__ frame_id,
                              float* __restrict__ out_oid, float* __restrict__ out_fid,
                              float* __restrict__ out_val) {
    const int s = blockIdx.x * blockDim.x + threadIdx.x;
    if (s >= CAP) return;
    const int j = inv[s];
    out_oid[s] = (float)((j >= 0) ? slot_ids[j] : obj_ids[s]);
    out_fid[s] = (float)((j >= 0) ? frame_id[0] : fid_buf[s]);
    out_val[s] = ((j >= 0) || (val_buf[s] != 0)) ? 1.0f : 0.0f;
}

// ------------- 1 GiB single-pass row copy via async global<->LDS DMA (CDNA5)
// Each block: 4 rows x 4KB. 256 lanes x B128 = one row per async burst.
// Data path: HBM -> LDS -> HBM through the async engine; VGPRs only hold
// addresses. ASYNCcnt synchronizes load completion before the LDS readback.
__global__ void row_copy_async(const float* __restrict__ tokens,
                               const float* __restrict__ memtok,
                               const int* __restrict__ inv,
                               float* __restrict__ out_mem) {
    __shared__ __align__(16) float buf[RPB * DM];            // 16 KB LDS
    const int t = threadIdx.x;
    // Low 32 bits of the generic pointer == LDS byte offset (aperture in high word).
    const unsigned lds_base = (unsigned)(size_t)(&buf[0]);

#pragma unroll
    for (int r = 0; r < RPB; ++r) {
        const int s = blockIdx.x * RPB + r;
        const int j = inv[s];
        const float* p = (j >= 0) ? (tokens + (size_t)j * DM)
                                  : (memtok + (size_t)s * DM);
        unsigned long long ga = (unsigned long long)(p + t * 4);
        unsigned lds = lds_base + (unsigned)(r * (DM * 4) + t * 16);
        asm volatile("global_load_async_to_lds_b128 %0, %1, off"
                     :: "v"(lds), "v"(ga) : "memory");
    }
    asm volatile("s_wait_asynccnt 0" ::: "memory");          // all 4 rows resident in LDS

#pragma unroll
    for (int r = 0; r < RPB; ++r) {
        const int s = blockIdx.x * RPB + r;
        unsigned long long gd = (unsigned long long)(out_mem + (size_t)s * DM + t * 4);
        unsigned lds = lds_base + (unsigned)(r * (DM * 4) + t * 16);
        asm volatile("global_store_async_from_lds_b128 %0, %1, off"
                     :: "v"(gd), "v"(lds) : "memory");
    }
    asm volatile("s_wait_asynccnt 0" ::: "memory");
}

// ---------------------------------------------------------------------- launch
extern "C" void kernel_launch(void* const* d_in, const int* in_sizes, int n_in,
                              void* d_out, int out_size, void* d_ws, size_t ws_size,
                              hipStream_t stream) {
    const float* tokens   = (const float*)d_in[0];
    const float* memtok   = (const float*)d_in[1];
    const int*   slot_ids = (const int*)d_in[2];
    const int*   vis      = (const int*)d_in[3];
    const int*   val      = (const int*)d_in[4];
    const int*   obj_ids  = (const int*)d_in[5];
    const int*   fid_buf  = (const int*)d_in[6];
    const int*   val_buf  = (const int*)d_in[7];
    const int*   frame_id = (const int*)d_in[8];
    const int*   wp       = (const int*)d_in[9];

    float* out      = (float*)d_out;
    float* out_mem  = out;                               // CAP*DM
    float* out_oid  = out_mem + (size_t)CAP * DM;        // CAP
    float* out_fid  = out_oid + CAP;                     // CAP
    float* out_val  = out_fid + CAP;                     // CAP
    float* out_np   = out_val + CAP;                     // 1

    // Workspace: inv map (CAP ints) + block counts/offsets (512 each)
    int* inv          = (int*)d_ws;
    int* blockCounts  = inv + CAP;
    int* blockOffsets = blockCounts + 512;

    init_inv<<<CAP / 256, 256, 0, stream>>>(inv);
    count_mask<<<NTOK / 256, 256, 0, stream>>>(vis, val, blockCounts);
    scan_block_counts<<<1, 512, 0, stream>>>(blockCounts, blockOffsets, wp, out_np);
    scatter_inv<<<NTOK / 256, 256, 0, stream>>>(vis, val, blockOffsets, wp, inv);
    small_outputs<<<CAP / 256, 256, 0, stream>>>(inv, slot_ids, obj_ids, fid_buf, val_buf,
                                                 frame_id, out_oid, out_fid, out_val);
    row_copy_async<<<CAP / RPB, 256, 0, stream>>>(tokens, memtok, inv, out_mem);
}